// GCNPerturb_22273700397228
// MI455X (gfx1250) — compile-verified
//
#include <hip/hip_runtime.h>
#include <hip/hip_bf16.h>

#define N_NODES 50000
#define N_EDGES 800000
#define NF 64
#define NC 10
#define NG 64

typedef __attribute__((ext_vector_type(2))) float v2f;
typedef __attribute__((ext_vector_type(8))) float v8f;

// ---------------- normalization pipeline ----------------

__global__ void k_zero(float* __restrict__ deg, float* __restrict__ pooled,
                       float* __restrict__ cnt) {
    int i = blockIdx.x * blockDim.x + threadIdx.x;
    if (i < N_NODES) deg[i] = 0.0f;
    if (i < NG * NF) pooled[i] = 0.0f;
    if (i < NG) cnt[i] = 0.0f;
}

__global__ void k_deg(const int* __restrict__ col, const float* __restrict__ P,
                      float* __restrict__ deg) {
    int e = blockIdx.x * blockDim.x + threadIdx.x;
    if (e >= N_EDGES) return;
    float w = 1.0f / (1.0f + expf(-P[e]));
    atomicAdd(&deg[col[e]], w);
}

__global__ void k_dis(float* __restrict__ deg_dis) {
    int i = blockIdx.x * blockDim.x + threadIdx.x;
    if (i >= N_NODES) return;
    float d = deg_dis[i] + 1.0f;  // + self-loop weight 1.0
    deg_dis[i] = (d > 0.0f) ? rsqrtf(d) : 0.0f;
}

__global__ void k_norm(const int* __restrict__ row, const int* __restrict__ col,
                       const float* __restrict__ P, const float* __restrict__ dis,
                       float* __restrict__ norm) {
    int e = blockIdx.x * blockDim.x + threadIdx.x;
    if (e >= N_EDGES) return;
    float w = 1.0f / (1.0f + expf(-P[e]));
    norm[e] = dis[row[e]] * w * dis[col[e]];
}

// ---------------- f32 WMMA GEMM: O[n,64] = H[n,64] @ W[64,64] ----------------
// block = 128 threads (4 wave32); each wave computes a 16x64 output slab.
// A tile (64x64) and W (64x64) staged into LDS via gfx1250 async DMA
// (global_load_async_to_lds_b128, ASYNCcnt) — no VGPR round-trip.
// Padded LDS strides give conflict-free access under the 16x16x4 f32 layout;
// both strides are multiples of 16B so b128 async chunks stay aligned.

#define AS_STRIDE 68   // 272B/row: banks 4m+k, conflict-free across half-waves
#define WS_STRIDE 80   // 320B/row: banks 16k+n, halves hit disjoint bank sets

__global__ void __launch_bounds__(128)
k_gemm64(const float* __restrict__ H, const float* __restrict__ Wg,
         float* __restrict__ O, int nrows) {
    __shared__ float Ws[64 * WS_STRIDE];
    __shared__ float As[64 * AS_STRIDE];
    const int tid  = threadIdx.x;
    const int base = blockIdx.x * 64;

    if (base + 64 <= nrows) {
        // Fast path: DMA 64x64 W and 64x64 A straight into LDS.
        // 1024 chunks of 16B each per matrix; 8 chunks/thread/matrix.
        for (int c = tid; c < 1024; c += 128) {
            int r = c >> 4, q = (c & 15) * 4;   // row, float offset in row
            unsigned wl = (unsigned)(uintptr_t)(&Ws[r * WS_STRIDE + q]);
            unsigned long long wg = (unsigned long long)(uintptr_t)(Wg + (c << 2));
            asm volatile("global_load_async_to_lds_b128 %0, %1, off"
                         :: "v"(wl), "v"(wg) : "memory");
            unsigned al = (unsigned)(uintptr_t)(&As[r * AS_STRIDE + q]);
            unsigned long long ag =
                (unsigned long long)(uintptr_t)(H + (size_t)(base + r) * 64 + q);
            asm volatile("global_load_async_to_lds_b128 %0, %1, off"
                         :: "v"(al), "v"(ag) : "memory");
        }
        asm volatile("s_wait_asynccnt 0x0" ::: "memory");
    } else {
        // Tail block: bounds-checked scalar staging with zero-fill.
        for (int idx = tid; idx < 64 * 64; idx += 128) {
            int r = idx >> 6, c = idx & 63;
            Ws[r * WS_STRIDE + c] = Wg[idx];
            int node = base + r;
            As[r * AS_STRIDE + c] = (node < nrows) ? H[node * 64 + c] : 0.0f;
        }
    }
    __syncthreads();

    const int wave = tid >> 5;
    const int lane = tid & 31;
    const int half = lane >> 4;   // 0: lanes 0-15, 1: lanes 16-31
    const int l16  = lane & 15;
    const int arow = wave * 16 + l16;   // A-matrix row (M) held by this lane
    const int nodeb = base + wave * 16;

    for (int t = 0; t < 4; ++t) {
        const int n0 = t * 16;
        v8f acc = {};
        #pragma unroll
        for (int k = 0; k < 64; k += 4) {
            // A 16x4 f32: VGPR0 = K(k+2h), VGPR1 = K(k+2h+1) per half-wave
            v2f a, b;
            a.x = As[arow * AS_STRIDE + k + 2 * half];
            a.y = As[arow * AS_STRIDE + k + 2 * half + 1];
            // B 4x16 f32: row K striped across lanes within a VGPR
            b.x = Ws[(k + 2 * half) * WS_STRIDE + n0 + l16];
            b.y = Ws[(k + 2 * half + 1) * WS_STRIDE + n0 + l16];
            acc = __builtin_amdgcn_wmma_f32_16x16x4_f32(
                false, a, false, b, (short)0, acc, false, false);
        }
        // C/D layout: VGPR r -> M=r (lanes 0-15), M=r+8 (lanes 16-31); N=l16
        #pragma unroll
        for (int r = 0; r < 8; ++r) {
            int node = nodeb + r + 8 * half;
            if (node < nrows) O[node * 64 + n0 + l16] = acc[r];
        }
    }
}

// ---------------- aggregation ----------------

// Agg[i,:] = dis[i]^2 * H[i,:]   (self-loop term; also initializes Agg)
__global__ void k_init_agg(const float* __restrict__ dis, const float* __restrict__ H,
                           float* __restrict__ Agg) {
    int idx = blockIdx.x * blockDim.x + threadIdx.x;
    if (idx >= N_NODES * NF) return;
    float d = dis[idx >> 6];
    Agg[idx] = d * d * H[idx];
}

// one wave per edge: Agg[col,:] += norm[e] * H[row,:]
__global__ void k_scatter(const int* __restrict__ row, const int* __restrict__ col,
                          const float* __restrict__ norm, const float* __restrict__ H,
                          float* __restrict__ Agg) {
    int warp = (blockIdx.x * blockDim.x + threadIdx.x) >> 5;
    int lane = threadIdx.x & 31;
    int nwarps = (gridDim.x * blockDim.x) >> 5;
    for (int e = warp; e < N_EDGES; e += nwarps) {
        int r = row[e];
        int c = col[e];
        float nv = norm[e];
        float h0 = H[r * 64 + lane];
        float h1 = H[r * 64 + 32 + lane];
        atomicAdd(&Agg[c * 64 + lane], nv * h0);
        atomicAdd(&Agg[c * 64 + 32 + lane], nv * h1);
    }
}

__global__ void k_bias_act(float* __restrict__ H, const float* __restrict__ b,
                           int relu) {
    int idx = blockIdx.x * blockDim.x + threadIdx.x;
    if (idx >= N_NODES * NF) return;
    float v = H[idx] + b[idx & 63];
    H[idx] = relu ? fmaxf(v, 0.0f) : v;
}

// ---------------- pooling + classifier ----------------

__global__ void k_pool(const int* __restrict__ batch, const float* __restrict__ H,
                       float* __restrict__ pooled, float* __restrict__ cnt) {
    int warp = (blockIdx.x * blockDim.x + threadIdx.x) >> 5;
    int lane = threadIdx.x & 31;
    int nwarps = (gridDim.x * blockDim.x) >> 5;
    for (int i = warp; i < N_NODES; i += nwarps) {
        int b = batch[i];
        atomicAdd(&pooled[b * 64 + lane], H[i * 64 + lane]);
        atomicAdd(&pooled[b * 64 + 32 + lane], H[i * 64 + 32 + lane]);
        if (lane == 0) atomicAdd(&cnt[b], 1.0f);
    }
}

__global__ void k_cls(const float* __restrict__ pooled, const float* __restrict__ cnt,
                      const float* __restrict__ Wl, const float* __restrict__ bl,
                      float* __restrict__ out) {
    int idx = blockIdx.x * blockDim.x + threadIdx.x;
    if (idx >= NG * NC) return;
    int g = idx / NC, c = idx % NC;
    float inv = 1.0f / fmaxf(cnt[g], 1.0f);
    float s = 0.0f;
    #pragma unroll
    for (int k = 0; k < 64; ++k)
        s = fmaf(pooled[g * 64 + k] * inv, Wl[k * NC + c], s);
    out[idx] = s + bl[c];
}

// ---------------- host launch ----------------

extern "C" void kernel_launch(void* const* d_in, const int* in_sizes, int n_in,
                              void* d_out, int out_size, void* d_ws, size_t ws_size,
                              hipStream_t stream) {
    const float* x     = (const float*)d_in[0];
    const int*   ei    = (const int*)d_in[1];
    const int*   row   = ei;             // edge_index[0]
    const int*   col   = ei + N_EDGES;   // edge_index[1]
    const int*   batch = (const int*)d_in[2];
    const float* P     = (const float*)d_in[3];
    const float* W[3]  = {(const float*)d_in[4], (const float*)d_in[6], (const float*)d_in[8]};
    const float* b[3]  = {(const float*)d_in[5], (const float*)d_in[7], (const float*)d_in[9]};
    const float* Wl    = (const float*)d_in[10];
    const float* bl    = (const float*)d_in[11];
    float* out = (float*)d_out;

    char* ws = (char*)d_ws;
    auto carve = [&](size_t nfloats) {
        float* p = (float*)ws;
        ws += ((nfloats * sizeof(float) + 255) / 256) * 256;
        return p;
    };
    float* dis    = carve(N_NODES);             // deg -> dis in place
    float* norm   = carve(N_EDGES);
    float* Hout   = carve((size_t)N_NODES * NF);
    float* Agg    = carve((size_t)N_NODES * NF);
    float* pooled = carve(NG * NF);
    float* cnt    = carve(NG);

    const int T = 256;
    const int gN  = (N_NODES + T - 1) / T;
    const int gE  = (N_EDGES + T - 1) / T;
    const int gNF = (N_NODES * NF + T - 1) / T;

    k_zero<<<gN, T, 0, stream>>>(dis, pooled, cnt);
    k_deg<<<gE, T, 0, stream>>>(col, P, dis);
    k_dis<<<gN, T, 0, stream>>>(dis);
    k_norm<<<gE, T, 0, stream>>>(row, col, P, dis, norm);

    const int gemmGrid = (N_NODES + 63) / 64;
    for (int l = 0; l < 3; ++l) {
        const float* Hin = (l == 0) ? x : Agg;
        k_gemm64<<<gemmGrid, 128, 0, stream>>>(Hin, W[l], Hout, N_NODES);
        k_init_agg<<<gNF, T, 0, stream>>>(dis, Hout, Agg);
        k_scatter<<<2048, T, 0, stream>>>(row, col, norm, Hout, Agg);
        k_bias_act<<<gNF, T, 0, stream>>>(Agg, b[l], l < 2 ? 1 : 0);
    }

    k_pool<<<1024, T, 0, stream>>>(batch, Agg, pooled, cnt);
    k_cls<<<(NG * NC + T - 1) / T, T, 0, stream>>>(pooled, cnt, Wl, bl, out);
}